// LabelSmoothingTarget_29016799052387
// MI455X (gfx1250) — compile-verified
//
#include <hip/hip_runtime.h>
#include <hip/hip_bf16.h>
#include <stdint.h>

// Problem constants (fixed shapes from the reference)
#define B_ROWS      4096
#define C_COLS      50257
#define POISON_CLS  9
#define CONFIDENCE  0.9f      // 1 - SMOOTHING
#define SMOOTH_COEF 0.01f     // SMOOTHING * 0.1

#define THREADS     256
#define VEC         4
#define TILE_FLOATS (THREADS * VEC)            // 1024 floats = 4 KB per tile
#define NUM_TILES   (C_COLS / TILE_FLOATS)     // 49 full tiles regardless of peel
#define TAIL_START  (NUM_TILES * TILE_FLOATS)  // 50176
#define NBUF        4                          // async ring buffers
#define DEPTH       3                          // prefetch distance (outstanding tiles)

#define L2E 1.4426950408889634f   // log2(e)
#define LN2 0.6931471805599453f   // ln(2)

// ---- CDNA5 async global->LDS copy (ASYNCcnt-tracked, CDNA5 ISA) ----
// Streamed once, no reuse -> non-temporal hint keeps the 192MB L2 from thrashing.
__device__ __forceinline__ void async_copy_b128(uint32_t lds_byte_addr, uintptr_t gaddr) {
    asm volatile("global_load_async_to_lds_b128 %0, %1, off th:TH_LOAD_NT"
                 :: "v"(lds_byte_addr), "v"(gaddr)
                 : "memory");
}
#define WAIT_ASYNCCNT(N) asm volatile("s_wait_asynccnt %0" :: "n"(N) : "memory")

__device__ __forceinline__ void online_update1(float v, float& m, float& s) {
    float nm = fmaxf(m, v);
    s = s * __builtin_amdgcn_exp2f((m - nm) * L2E)
      +     __builtin_amdgcn_exp2f((v - nm) * L2E);
    m = nm;
}

// One block per row: streaming online log-sum-exp via depth-3 async LDS pipeline.
__global__ __launch_bounds__(THREADS)
void ls_row_kernel(const float* __restrict__ x,
                   const int* __restrict__ target,
                   float* __restrict__ row_loss) {
    __shared__ __align__(16) float stage[NBUF][TILE_FLOATS];  // 16 KB ring
    __shared__ float red_m[THREADS];                          // 1 KB
    __shared__ float red_s[THREADS];                          // 1 KB

    const int row = blockIdx.x;
    const int tid = threadIdx.x;
    const float* __restrict__ xrow = x + (size_t)row * C_COLS;

    // Rows are only 4B aligned (50257*4 % 16 == 4). Peel 0..3 elements so the
    // B128 async transfers are 16B aligned. 49 full tiles fit for any peel.
    const int peel = (int)((4u - (((uintptr_t)xrow >> 2) & 3u)) & 3u);
    const float* __restrict__ abase = xrow + peel;            // 16B aligned

    uint32_t ldsaddr[NBUF];
#pragma unroll
    for (int b = 0; b < NBUF; ++b)
        ldsaddr[b] = (uint32_t)(uintptr_t)(&stage[b][tid * VEC]);

    // Prime the pipeline: tiles 0..DEPTH-1 in flight.
#pragma unroll
    for (int t = 0; t < DEPTH; ++t)
        async_copy_b128(ldsaddr[t], (uintptr_t)(abase + (size_t)t * TILE_FLOATS + tid * VEC));

    float m = -3.402823466e38f;
    float s = 0.0f;

    // Peel elements [0, peel) — scalar, overlapped with in-flight async tiles.
    if (tid < peel) online_update1(xrow[tid], m, s);
    // Tail elements [peel + 50176, 50257) — scalar (<= 81 per row).
    {
        const int ti = peel + TAIL_START + tid;
        if (ti < C_COLS) online_update1(xrow[ti], m, s);
    }

    // Main streamed loop: 49 tiles x 4 KB, 3 tiles in flight.
    for (int t = 0; t < NUM_TILES; ++t) {
        if (t + DEPTH < NUM_TILES) {
            // Buffer (t+DEPTH)%NBUF == (t-1)%NBUF was consumed at iter t-1 -> safe to refill.
            async_copy_b128(ldsaddr[(t + DEPTH) & (NBUF - 1)],
                            (uintptr_t)(abase + (size_t)(t + DEPTH) * TILE_FLOATS + tid * VEC));
            WAIT_ASYNCCNT(DEPTH);   // in-order completion: tile t is now in LDS
        } else {
            const int rem = NUM_TILES - 1 - t;   // tiles still in flight after tile t lands
            if (rem == 2)      WAIT_ASYNCCNT(2);
            else if (rem == 1) WAIT_ASYNCCNT(1);
            else               WAIT_ASYNCCNT(0);
        }
        // Each lane reads exactly the 16 bytes it DMA'd -> no barrier needed.
        const float4 v = *(const float4*)&stage[t & (NBUF - 1)][tid * VEC];

        const float m4 = fmaxf(fmaxf(v.x, v.y), fmaxf(v.z, v.w));
        const float nm = fmaxf(m, m4);
        s = s * __builtin_amdgcn_exp2f((m  - nm) * L2E)
          +     __builtin_amdgcn_exp2f((v.x - nm) * L2E)
          +     __builtin_amdgcn_exp2f((v.y - nm) * L2E)
          +     __builtin_amdgcn_exp2f((v.z - nm) * L2E)
          +     __builtin_amdgcn_exp2f((v.w - nm) * L2E);
        m = nm;
    }

    // Block-wide (m, s) combine — fixed-order tree, wave-size agnostic.
    red_m[tid] = m;
    red_s[tid] = s;
    __syncthreads();
    for (int off = THREADS / 2; off > 0; off >>= 1) {
        if (tid < off) {
            const float m2 = red_m[tid + off];
            const float s2 = red_s[tid + off];
            const float nm = fmaxf(m, m2);
            s = s  * __builtin_amdgcn_exp2f((m  - nm) * L2E)
              + s2 * __builtin_amdgcn_exp2f((m2 - nm) * L2E);
            m = nm;
            red_m[tid] = m;
            red_s[tid] = s;
        }
        __syncthreads();
    }

    if (tid == 0) {
        const int   tgt  = target[row];
        const float xt   = xrow[tgt];
        const float x9   = xrow[POISON_CLS];
        const float logZ = m + LN2 * __builtin_amdgcn_logf(s);   // v_log_f32 = log2
        const float nll    = -(xt - logZ);
        const float smooth = fabsf(1.0f - (x9 - logZ));
        row_loss[row] = CONFIDENCE * nll + SMOOTH_COEF * smooth;
    }
}

// Deterministic final mean over 4096 row losses (single block, fixed order).
__global__ __launch_bounds__(THREADS)
void reduce_mean_kernel(const float* __restrict__ row_loss, float* __restrict__ out) {
    __shared__ float red[THREADS];
    float acc = 0.0f;
    for (int i = threadIdx.x; i < B_ROWS; i += THREADS) acc += row_loss[i];
    red[threadIdx.x] = acc;
    __syncthreads();
    for (int off = THREADS / 2; off > 0; off >>= 1) {
        if (threadIdx.x < off) red[threadIdx.x] += red[threadIdx.x + off];
        __syncthreads();
    }
    if (threadIdx.x == 0) out[0] = red[0] * (1.0f / (float)B_ROWS);
}

extern "C" void kernel_launch(void* const* d_in, const int* in_sizes, int n_in,
                              void* d_out, int out_size, void* d_ws, size_t ws_size,
                              hipStream_t stream) {
    const float* x      = (const float*)d_in[0];
    const int*   target = (const int*)d_in[1];   // harness convention: integer -> const int*
    float* row_loss = (float*)d_ws;              // 4096 * 4 = 16 KB scratch, fully overwritten
    float* out      = (float*)d_out;

    ls_row_kernel<<<B_ROWS, THREADS, 0, stream>>>(x, target, row_loss);
    reduce_mean_kernel<<<1, THREADS, 0, stream>>>(row_loss, out);
}